// HyperGraphModule_1357209665997
// MI455X (gfx1250) — compile-verified
//
#include <hip/hip_runtime.h>
#include <math.h>

// ---------------------------------------------------------------------------
// Hypergraph double-convolution, algebraically rewritten:
//   segment_sum(c_e * (x[s_e] @ W^T + b))
//     = (segment_sum(c_e * x[s_e])) @ W^T + segment_sum(c_e) * b
// Phase 1: per-edge scaled scatter-add into per-receiver accumulators (L2-resident)
// Phase 2: dense 16x16-tiled GEMM on accumulators via V_WMMA_F32_16X16X4_F32
//          with fused rank-1 bias + tanh(gscale*gmsg) epilogue.
// ---------------------------------------------------------------------------

typedef float v2f __attribute__((ext_vector_type(2)));
typedef float v8f __attribute__((ext_vector_type(8)));

#define FD 128  // feature dimension

// One wave (32 lanes) per edge; lane i handles float4 chunk i of the 128-f32 row.
// Fully coalesced 512B row load + 512B of f32 atomics into the L2-resident acc.
__global__ void scatter_scale_add_kernel(const float* __restrict__ feat,
                                         const float* __restrict__ conv,
                                         const int*   __restrict__ senders,
                                         const int*   __restrict__ receivers,
                                         float* __restrict__ acc,
                                         float* __restrict__ csum,
                                         int nEdges) {
    long long tid = (long long)blockIdx.x * blockDim.x + threadIdx.x;
    int e    = (int)(tid >> 5);
    int lane = (int)(tid & 31);
    if (e >= nEdges) return;

    int   s = senders[e];
    int   r = receivers[e];
    float c = conv[e];

    float4 v = ((const float4*)(feat + (size_t)s * FD))[lane];
    float* dst = acc + (size_t)r * FD + (size_t)lane * 4;
    atomicAdd(dst + 0, c * v.x);
    atomicAdd(dst + 1, c * v.y);
    atomicAdd(dst + 2, c * v.z);
    atomicAdd(dst + 3, c * v.w);
    if (lane == 0) atomicAdd(csum + r, c);
}

// Block = 256 threads = 8 waves. Block b owns rows [16b, 16b+16); wave w owns
// cols [16w, 16w+16) (FD=128 -> 8 col tiles). Each wave runs two K=128 GEMM
// accumulations (msg and scale weights) as 2x32 V_WMMA_F32_16X16X4_F32, then
// applies the rank-1 bias terms and the fused tanh epilogue.
//
// WMMA f32 16x16x4 lane layouts (ISA 7.12.2):
//   A: lane L -> a = { A[L%16][2*(L/16)], A[L%16][2*(L/16)+1] }
//   B: lane L -> b = { B[2*(L/16)][L%16], B[2*(L/16)+1][L%16] }, B[k][n]=W[n][k]
//   C/D: VGPR v, lane L -> element [v + 8*(L/16)][L%16]
__global__ void fused_gemm_bias_tanh_kernel(const float* __restrict__ accMsg,
                                            const float* __restrict__ cMsg,
                                            const float* __restrict__ Wmsg,
                                            const float* __restrict__ bMsg,
                                            const float* __restrict__ accScale,
                                            const float* __restrict__ cScale,
                                            const float* __restrict__ Wscale,
                                            const float* __restrict__ bScale,
                                            float* __restrict__ out,
                                            int nRows) {
    int wave = threadIdx.x >> 5;
    int lane = threadIdx.x & 31;
    int rowBase = blockIdx.x * 16;
    int colBase = wave * 16;

    int mA   = lane & 15;          // A-matrix row within tile for this lane
    int kGrp = (lane >> 4) * 2;    // k-pair offset within each K=4 step
    int nCol = lane & 15;          // B/C/D column within tile

    const float* pA1 = accMsg   + (size_t)(rowBase + mA)  * FD + kGrp;
    const float* pB1 = Wmsg     + (size_t)(colBase + nCol) * FD + kGrp;
    const float* pA2 = accScale + (size_t)(rowBase + mA)  * FD + kGrp;
    const float* pB2 = Wscale   + (size_t)(colBase + nCol) * FD + kGrp;

    v8f c1 = {};
    v8f c2 = {};
#pragma unroll
    for (int k0 = 0; k0 < FD; k0 += 4) {
        v2f a1 = *(const v2f*)(pA1 + k0);
        v2f b1 = *(const v2f*)(pB1 + k0);
        c1 = __builtin_amdgcn_wmma_f32_16x16x4_f32(false, a1, false, b1,
                                                   (short)0, c1, false, false);
        v2f a2 = *(const v2f*)(pA2 + k0);
        v2f b2 = *(const v2f*)(pB2 + k0);
        c2 = __builtin_amdgcn_wmma_f32_16x16x4_f32(false, a2, false, b2,
                                                   (short)0, c2, false, false);
    }

    int   n  = colBase + nCol;
    float bm = bMsg[n];
    float bs = bScale[n];
    int rowOff = (lane >> 4) * 8;
#pragma unroll
    for (int v = 0; v < 8; ++v) {
        int m = rowBase + rowOff + v;
        if (m < nRows) {
            float gMsg   = c1[v] + cMsg[m]   * bm;   // acc@W^T + sum(c)*b
            float gScale = c2[v] + cScale[m] * bs;
            out[(size_t)m * FD + n] = tanhf(gScale * gMsg);
        }
    }
}

extern "C" void kernel_launch(void* const* d_in, const int* in_sizes, int n_in,
                              void* d_out, int out_size, void* d_ws, size_t ws_size,
                              hipStream_t stream) {
    (void)n_in; (void)out_size; (void)ws_size;

    // Inputs in setup_inputs() order
    const float* node_features  = (const float*)d_in[0];
    const float* hedge_features = (const float*)d_in[1];
    const float* W_node_msg     = (const float*)d_in[2];
    const float* b_node_msg     = (const float*)d_in[3];
    const float* W_hedge_scale  = (const float*)d_in[4];
    const float* b_hedge_scale  = (const float*)d_in[5];
    const float* W_hedge_msg    = (const float*)d_in[6];
    const float* b_hedge_msg    = (const float*)d_in[7];
    const float* W_node_scale   = (const float*)d_in[8];
    const float* b_node_scale   = (const float*)d_in[9];
    const float* node_conv      = (const float*)d_in[10];
    const float* h2n_conv       = (const float*)d_in[11];
    const float* hedge_conv     = (const float*)d_in[12];
    const float* n2h_conv       = (const float*)d_in[13];
    const int*   node_senders   = (const int*)d_in[14];
    const int*   node_receivers = (const int*)d_in[15];
    const int*   h2n_senders    = (const int*)d_in[16];
    const int*   h2n_receivers  = (const int*)d_in[17];
    const int*   hedge_senders  = (const int*)d_in[18];
    const int*   hedge_receivers= (const int*)d_in[19];
    const int*   n2h_senders    = (const int*)d_in[20];
    const int*   n2h_receivers  = (const int*)d_in[21];

    const int nNodes  = in_sizes[0] / FD;
    const int nHedges = in_sizes[1] / FD;
    const int eNN     = in_sizes[10];
    const int eINC    = in_sizes[11];
    const int eHH     = in_sizes[12];

    // Pad row counts to WMMA tile (16) so tile loads are always in-bounds.
    const int nodePad  = ((nNodes  + 15) / 16) * 16;
    const int hedgePad = ((nHedges + 15) / 16) * 16;

    // Workspace layout (floats)
    float* ws = (float*)d_ws;
    size_t off = 0;
    float* accA = ws + off; off += (size_t)nodePad  * FD;  // node msg acc
    float* accB = ws + off; off += (size_t)nodePad  * FD;  // node scale acc
    float* accC = ws + off; off += (size_t)hedgePad * FD;  // hedge msg acc
    float* accD = ws + off; off += (size_t)hedgePad * FD;  // hedge scale acc
    float* cA   = ws + off; off += (size_t)nodePad;
    float* cB   = ws + off; off += (size_t)nodePad;
    float* cC   = ws + off; off += (size_t)hedgePad;
    float* cD   = ws + off; off += (size_t)hedgePad;

    hipMemsetAsync(d_ws, 0, off * sizeof(float), stream);

    float* out_nodes  = (float*)d_out;                         // [nNodes, 128]
    float* out_hedges = (float*)d_out + (size_t)nNodes * FD;   // [nHedges, 128]

    const int TPB = 256;
    auto blocksFor = [](long long edges) {
        return (int)((edges * 32 + 255) / 256);
    };

    // --- Phase 1 scatters (independent of node update) ---
    scatter_scale_add_kernel<<<blocksFor(eNN), TPB, 0, stream>>>(
        node_features, node_conv, node_senders, node_receivers, accA, cA, eNN);
    scatter_scale_add_kernel<<<blocksFor(eINC), TPB, 0, stream>>>(
        hedge_features, h2n_conv, h2n_senders, h2n_receivers, accB, cB, eINC);
    scatter_scale_add_kernel<<<blocksFor(eHH), TPB, 0, stream>>>(
        hedge_features, hedge_conv, hedge_senders, hedge_receivers, accC, cC, eHH);

    // --- Node update: tanh((accB@Whs^T + cB*bhs) * (accA@Wnm^T + cA*bnm)) ---
    fused_gemm_bias_tanh_kernel<<<nodePad / 16, TPB, 0, stream>>>(
        accA, cA, W_node_msg, b_node_msg,
        accB, cB, W_hedge_scale, b_hedge_scale,
        out_nodes, nNodes);

    // --- node2hedge scatter uses the UPDATED node features ---
    scatter_scale_add_kernel<<<blocksFor(eINC), TPB, 0, stream>>>(
        out_nodes, n2h_conv, n2h_senders, n2h_receivers, accD, cD, eINC);

    // --- Hedge update: tanh((accD@Wns^T + cD*bns) * (accC@Whm^T + cC*bhm)) ---
    fused_gemm_bias_tanh_kernel<<<hedgePad / 16, TPB, 0, stream>>>(
        accC, cC, W_hedge_msg, b_hedge_msg,
        accD, cD, W_node_scale, b_node_scale,
        out_hedges, nHedges);
}